// HeteroGNN_75428215652543
// MI455X (gfx1250) — compile-verified
//
#include <hip/hip_runtime.h>
#include <hip/hip_bf16.h>

typedef __attribute__((ext_vector_type(2))) float v2f;
typedef __attribute__((ext_vector_type(8))) float v8f;

#define C 128
#define N_USER 50000
#define N_ITEM 50000

// ---------------------------------------------------------------------------
// Edge scatter: one wave (32 lanes) per edge. Each lane moves 4 floats.
// agg[dst] += x_src[src]; cnt[dst] += 1  (f32 atomics, L2-resident targets)
// ---------------------------------------------------------------------------
__global__ __launch_bounds__(256)
void scatter_add_kernel(const float* __restrict__ xsrc,
                        const int* __restrict__ src,
                        const int* __restrict__ dst,
                        float* __restrict__ agg,
                        float* __restrict__ cnt,
                        int nE) {
    int wid  = (blockIdx.x * blockDim.x + threadIdx.x) >> 5;
    int lane = threadIdx.x & 31;
    if (wid >= nE) return;
    int s = src[wid];
    int d = dst[wid];
    const float4 v = *(const float4*)(xsrc + (size_t)s * C + lane * 4);
    float* p = agg + (size_t)d * C + lane * 4;
    atomicAdd(p + 0, v.x);
    atomicAdd(p + 1, v.y);
    atomicAdd(p + 2, v.z);
    atomicAdd(p + 3, v.w);
    if (lane == 0) atomicAdd(cnt + d, 1.0f);
}

// ---------------------------------------------------------------------------
// WMMA fragment helpers (V_WMMA_F32_16X16X4_F32 layouts, ISA 7.12.2)
//   A 16x4:  lane m(0..15): VGPR0=K0,VGPR1=K1 ; lane m+16: K2,K3
//   B 4x16:  lane n(0..15): VGPR0=K0,VGPR1=K1 ; lane n+16: K2,K3
//   C/D 16x16: VGPR r -> row r (lanes 0-15) / row r+8 (lanes 16-31), col=lane&15
// ---------------------------------------------------------------------------

// Item-side: out = mean_ui @ Wl.T + b + x_item @ Wr.T   (in-place over aggUI)
__global__ __launch_bounds__(256)
void gemm_item_kernel(float* __restrict__ aggUI,          // [N,C] in: sums, out: pre-LN
                      const float* __restrict__ xdst,     // [N,C]
                      const float* __restrict__ cntUI,    // [N]
                      const float* __restrict__ Wl,       // [C,C] row-major (out x in)
                      const float* __restrict__ Wr,
                      const float* __restrict__ bias) {
    __shared__ float sA1[16][C];
    __shared__ float sA2[16][C];
    __shared__ float scale1[16];
    const int r0 = blockIdx.x * 16;
    const int t  = threadIdx.x;
    if (t < 16) scale1[t] = 1.0f / fmaxf(cntUI[r0 + t], 1.0f);
    __syncthreads();
    for (int idx = t; idx < 16 * C; idx += 256) {
        int r = idx >> 7, c = idx & (C - 1);
        sA1[r][c] = aggUI[(size_t)(r0 + r) * C + c] * scale1[r];
        sA2[r][c] = xdst[(size_t)(r0 + r) * C + c];
    }
    __syncthreads();

    const int lane = t & 31;
    const int c0   = (t >> 5) * 16;       // this wave's 16 output columns
    const int m    = lane & 15;
    const int kb   = (lane >> 4) << 1;    // 0 or 2
    const int n    = lane & 15;
    const float bv = bias[c0 + n];
    v8f acc = {bv, bv, bv, bv, bv, bv, bv, bv};

    for (int k0 = 0; k0 < C; k0 += 4) {
        const int kk = k0 + kb;
        v2f a1 = {sA1[m][kk], sA1[m][kk + 1]};
        v2f a2 = {sA2[m][kk], sA2[m][kk + 1]};
        v2f b1 = *(const v2f*)(Wl + (size_t)(c0 + n) * C + kk);
        v2f b2 = *(const v2f*)(Wr + (size_t)(c0 + n) * C + kk);
        acc = __builtin_amdgcn_wmma_f32_16x16x4_f32(false, a1, false, b1, (short)0, acc, false, false);
        acc = __builtin_amdgcn_wmma_f32_16x16x4_f32(false, a2, false, b2, (short)0, acc, false, false);
    }

    const int mbase = (lane >> 4) << 3;   // 0 or 8
#pragma unroll
    for (int r = 0; r < 8; ++r)
        aggUI[(size_t)(r0 + mbase + r) * C + c0 + n] = acc[r];
}

// User-side (HeteroConv sum of two SAGE branches):
// out = mean_iu@Wl_iu.T + mean_uu@Wl_uu.T + x_u@(Wr_iu+Wr_uu).T + b_iu + b_uu
// written in-place over aggIU.
__global__ __launch_bounds__(256)
void gemm_user_kernel(float* __restrict__ aggIU,
                      const float* __restrict__ aggUU,
                      const float* __restrict__ xu,
                      const float* __restrict__ cntIU,
                      const float* __restrict__ cntUU,
                      const float* __restrict__ Wl_iu,
                      const float* __restrict__ Wr_iu,
                      const float* __restrict__ Wl_uu,
                      const float* __restrict__ Wr_uu,
                      const float* __restrict__ b_iu,
                      const float* __restrict__ b_uu) {
    __shared__ float sA1[16][C];
    __shared__ float sA2[16][C];
    __shared__ float sA3[16][C];
    __shared__ float scale1[16];
    __shared__ float scale2[16];
    const int r0 = blockIdx.x * 16;
    const int t  = threadIdx.x;
    if (t < 16) {
        scale1[t] = 1.0f / fmaxf(cntIU[r0 + t], 1.0f);
        scale2[t] = 1.0f / fmaxf(cntUU[r0 + t], 1.0f);
    }
    __syncthreads();
    for (int idx = t; idx < 16 * C; idx += 256) {
        int r = idx >> 7, c = idx & (C - 1);
        sA1[r][c] = aggIU[(size_t)(r0 + r) * C + c] * scale1[r];
        sA2[r][c] = aggUU[(size_t)(r0 + r) * C + c] * scale2[r];
        sA3[r][c] = xu[(size_t)(r0 + r) * C + c];
    }
    __syncthreads();

    const int lane = t & 31;
    const int c0   = (t >> 5) * 16;
    const int m    = lane & 15;
    const int kb   = (lane >> 4) << 1;
    const int n    = lane & 15;
    const float bv = b_iu[c0 + n] + b_uu[c0 + n];
    v8f acc = {bv, bv, bv, bv, bv, bv, bv, bv};

    for (int k0 = 0; k0 < C; k0 += 4) {
        const int kk = k0 + kb;
        v2f a1 = {sA1[m][kk], sA1[m][kk + 1]};
        v2f a2 = {sA2[m][kk], sA2[m][kk + 1]};
        v2f a3 = {sA3[m][kk], sA3[m][kk + 1]};
        v2f b1 = *(const v2f*)(Wl_iu + (size_t)(c0 + n) * C + kk);
        v2f b2 = *(const v2f*)(Wl_uu + (size_t)(c0 + n) * C + kk);
        v2f br = *(const v2f*)(Wr_iu + (size_t)(c0 + n) * C + kk)
               + *(const v2f*)(Wr_uu + (size_t)(c0 + n) * C + kk);
        acc = __builtin_amdgcn_wmma_f32_16x16x4_f32(false, a1, false, b1, (short)0, acc, false, false);
        acc = __builtin_amdgcn_wmma_f32_16x16x4_f32(false, a2, false, b2, (short)0, acc, false, false);
        acc = __builtin_amdgcn_wmma_f32_16x16x4_f32(false, a3, false, br, (short)0, acc, false, false);
    }

    const int mbase = (lane >> 4) << 3;
#pragma unroll
    for (int r = 0; r < 8; ++r)
        aggIU[(size_t)(r0 + mbase + r) * C + c0 + n] = acc[r];
}

// ---------------------------------------------------------------------------
// LayerNorm (per-node over 128 features) + ReLU. 128 threads = 4 waves/node.
// ---------------------------------------------------------------------------
__global__ __launch_bounds__(128)
void ln_relu_kernel(const float* __restrict__ x,
                    const float* __restrict__ w,
                    const float* __restrict__ b,
                    float* __restrict__ out) {
    const int row = blockIdx.x;
    const int t   = threadIdx.x;
    __shared__ float red[4];
    float v = x[(size_t)row * C + t];

    float s = v;
    for (int off = 16; off; off >>= 1) s += __shfl_xor(s, off, 32);
    if ((t & 31) == 0) red[t >> 5] = s;
    __syncthreads();
    float mu = (red[0] + red[1] + red[2] + red[3]) * (1.0f / 128.0f);
    __syncthreads();

    float xc = v - mu;
    float q  = xc * xc;
    for (int off = 16; off; off >>= 1) q += __shfl_xor(q, off, 32);
    if ((t & 31) == 0) red[t >> 5] = q;
    __syncthreads();
    float var = (red[0] + red[1] + red[2] + red[3]) * (1.0f / 128.0f);

    float y = xc * rsqrtf(var + 1e-5f) * w[t] + b[t];
    out[(size_t)row * C + t] = fmaxf(y, 0.0f);
}

// ---------------------------------------------------------------------------
// Host-side orchestration
// ---------------------------------------------------------------------------
extern "C" void kernel_launch(void* const* d_in, const int* in_sizes, int n_in,
                              void* d_out, int out_size, void* d_ws, size_t ws_size,
                              hipStream_t stream) {
    const float* x_user = (const float*)d_in[0];
    const float* x_item = (const float*)d_in[1];
    const int*   ei_ui  = (const int*)d_in[2];
    const int*   ei_iu  = (const int*)d_in[3];
    const int*   ei_uu  = (const int*)d_in[4];
    const int E = in_sizes[2] / 2;

    // weights per layer, base index 5 (layer0) / 18 (layer1)
    auto P = [&](int i) { return (const float*)d_in[i]; };

    float* ws     = (float*)d_ws;
    float* aggUI  = ws;                                   // N_ITEM*C
    float* aggIU  = ws + (size_t)N_ITEM * C;              // N_USER*C
    float* aggUU  = aggIU + (size_t)N_USER * C;           // N_USER*C
    float* cntUI  = aggUU + (size_t)N_USER * C;           // N_ITEM
    float* cntIU  = cntUI + N_ITEM;                       // N_USER
    float* cntUU  = cntIU + N_USER;                       // N_USER
    const size_t ws_used = ((size_t)N_ITEM * C + 2 * (size_t)N_USER * C +
                            N_ITEM + 2 * N_USER) * sizeof(float);

    float* out_xu = (float*)d_out;                        // [N_USER, C]
    float* out_xi = out_xu + (size_t)N_USER * C;          // [N_ITEM, C]

    const int scatter_blocks = (E + 7) / 8;               // 8 waves/block, 1 edge/wave
    const int gemm_u_blocks  = N_USER / 16;
    const int gemm_i_blocks  = N_ITEM / 16;

    const float* xu = x_user;
    const float* xi = x_item;

    for (int l = 0; l < 2; ++l) {
        const int base = 5 + l * 13;
        const float *Wl_ui = P(base + 0), *b_ui = P(base + 1), *Wr_ui = P(base + 2);
        const float *Wl_iu = P(base + 3), *b_iu = P(base + 4), *Wr_iu = P(base + 5);
        const float *Wl_uu = P(base + 6), *b_uu = P(base + 7), *Wr_uu = P(base + 8);
        const float *lnw_u = P(base + 9), *lnb_u = P(base + 10);
        const float *lnw_i = P(base + 11), *lnb_i = P(base + 12);

        hipMemsetAsync(d_ws, 0, ws_used, stream);

        scatter_add_kernel<<<scatter_blocks, 256, 0, stream>>>(xu, ei_ui, ei_ui + E, aggUI, cntUI, E);
        scatter_add_kernel<<<scatter_blocks, 256, 0, stream>>>(xi, ei_iu, ei_iu + E, aggIU, cntIU, E);
        scatter_add_kernel<<<scatter_blocks, 256, 0, stream>>>(xu, ei_uu, ei_uu + E, aggUU, cntUU, E);

        gemm_item_kernel<<<gemm_i_blocks, 256, 0, stream>>>(aggUI, xi, cntUI, Wl_ui, Wr_ui, b_ui);
        gemm_user_kernel<<<gemm_u_blocks, 256, 0, stream>>>(aggIU, aggUU, xu, cntIU, cntUU,
                                                            Wl_iu, Wr_iu, Wl_uu, Wr_uu, b_iu, b_uu);

        ln_relu_kernel<<<N_USER, 128, 0, stream>>>(aggIU, lnw_u, lnb_u, out_xu);
        ln_relu_kernel<<<N_ITEM, 128, 0, stream>>>(aggUI, lnw_i, lnb_i, out_xi);

        xu = out_xu;   // layer-2 inputs come from d_out; LN overwrites them last
        xi = out_xi;
    }
}